// ViterbiDecoder_74448963109189
// MI455X (gfx1250) — compile-verified
//
#include <hip/hip_runtime.h>
#include <stdint.h>

// Soft-Viterbi DP, single workgroup (32 wave32 waves = 1 WGP), row-parallel with
// log-domain linear-recurrence scan for the y-state. Theta rows are streamed
// Global->LDS with the CDNA5 Tensor Data Mover (async, TENSORcnt), double-buffered
// two rows ahead of compute.

#define NROWS 2048
#define MCOLS 2048
#define ROWELEMS (MCOLS * 3)   // 6144 floats per theta row

typedef unsigned int v4u __attribute__((ext_vector_type(4)));
typedef int          v8i __attribute__((ext_vector_type(8)));
typedef int          v4i __attribute__((ext_vector_type(4)));

__device__ __forceinline__ float lse2(float a, float b) {
  float m = fmaxf(a, b);
  return m + __logf(__expf(a - m) + __expf(b - m));
}
__device__ __forceinline__ float lse3(float a, float b, float c) {
  float m = fmaxf(fmaxf(a, b), c);
  return m + __logf(__expf(a - m) + __expf(b - m) + __expf(c - m));
}
// logaddexp that tolerates -inf identity elements of the scan monoid
__device__ __forceinline__ float laddexp(float a, float b) {
  float m = fmaxf(a, b);
  if (!(m > -INFINITY)) return -INFINITY;
  return m + __logf(__expf(a - m) + __expf(b - m));
}

// TDM: 1-line tile of ROWELEMS f32 elements, Global -> LDS.
// D# layout per CDNA5 ISA.
__device__ __forceinline__ void tdm_load_row(const float* gsrc, unsigned lds_byte_off) {
  unsigned long long ga = (unsigned long long)(uintptr_t)gsrc;
  v4u g0;
  g0[0] = 1u;                                   // count=1, user descriptor
  g0[1] = lds_byte_off;                         // lds_addr
  g0[2] = (unsigned)(ga & 0xFFFFFFFFull);       // global_addr[31:0]
  g0[3] = (unsigned)((ga >> 32) & 0x01FFFFFFull) | 0x80000000u; // ga[56:32] | type=2
  v8i g1;
  g1[0] = 0x00020000;                           // data_size=2 (4 bytes), wg_mask=0
  g1[1] = (int)((ROWELEMS & 0xFFFF) << 16);     // tensor_dim0[15:0]
  g1[2] = (int)(1u << 16);                      // tensor_dim0[31:16]=0 | tensor_dim1=1
  g1[3] = (int)((ROWELEMS & 0xFFFF) << 16);     // tile_dim0 = 6144
  g1[4] = 1;                                    // tile_dim1 = 1, tile_dim2 = 0
  g1[5] = ROWELEMS;                             // tensor_dim0_stride
  g1[6] = 0;
  g1[7] = 0;
  v4i z4; z4[0] = 0; z4[1] = 0; z4[2] = 0; z4[3] = 0;
  v8i z8; z8[0] = 0; z8[1] = 0; z8[2] = 0; z8[3] = 0;
  z8[4] = 0; z8[5] = 0; z8[6] = 0; z8[7] = 0;
  // 6-arg form on this toolchain: (g0, g1, g2, g3, <second-descriptor slot>, cpol)
  __builtin_amdgcn_tensor_load_to_lds(g0, g1, z4, z4, z8, 0);
}

__global__ __launch_bounds__(1024)
void viterbi_soft_kernel(const float* __restrict__ theta,
                         const float* __restrict__ A,
                         float* __restrict__ out) {
  __shared__ float sV[(MCOLS + 1) * 3];      // 24588 B  current/prev V row (in place)
  __shared__ float sTheta[2][ROWELEMS];      // 49152 B  double-buffered theta rows
  __shared__ float sWg[32], sWc[32];         // wave scan totals
  __shared__ float sWgI[32], sWcI[32];       // inclusive wave prefixes

  const int tid  = threadIdx.x;
  const int lane = tid & 31;
  const int wid  = tid >> 5;
  const int jj0  = 2 * tid;
  const int jj1  = 2 * tid + 1;

  // Transition rows (state order x=0, m=1, y=2 per reference stack order)
  const float Ax0 = A[0], Ax1 = A[1], Ax2 = A[2];   // A[X_ST]
  const float Am0 = A[3], Am1 = A[4], Am2 = A[5];   // A[M_ST]
  const float Ay0 = A[6], Ay1 = A[7], Ay2 = A[8];   // A[Y_ST]

  // V0 row: zeros except V0[0][m] = 1
  sV[jj0 * 3 + 0] = 0.0f; sV[jj0 * 3 + 1] = 0.0f; sV[jj0 * 3 + 2] = 0.0f;
  sV[jj1 * 3 + 0] = 0.0f; sV[jj1 * 3 + 1] = 0.0f; sV[jj1 * 3 + 2] = 0.0f;
  if (tid == 0) {
    sV[MCOLS * 3 + 0] = 0.0f; sV[MCOLS * 3 + 1] = 0.0f; sV[MCOLS * 3 + 2] = 0.0f;
    sV[1] = 1.0f;
  }

  // Wave 0 prefetches theta rows 0 and 1 via TDM (EXEC-independent, once per wave)
  if (tid < 32) {
    tdm_load_row(theta, (unsigned)(uintptr_t)&sTheta[0][0]);
    tdm_load_row(theta + (size_t)ROWELEMS, (unsigned)(uintptr_t)&sTheta[1][0]);
  }

  for (int r = 0; r < NROWS; ++r) {
    // Row r's DMA complete when <=1 outstanding (in-order per wave); last row: 0.
    if (r == NROWS - 1) __builtin_amdgcn_s_wait_tensorcnt(0);
    else                __builtin_amdgcn_s_wait_tensorcnt(1);
    __syncthreads();  // B0: theta[r] in LDS; previous row's V writes visible

    const float* th = sTheta[r & 1];
    const float thx0 = th[jj0 * 3 + 0], thm0 = th[jj0 * 3 + 1], thy0 = th[jj0 * 3 + 2];
    const float thx1 = th[jj1 * 3 + 0], thm1 = th[jj1 * 3 + 1], thy1 = th[jj1 * 3 + 2];

    const float p0x = sV[jj0 * 3 + 0], p0m = sV[jj0 * 3 + 1], p0y = sV[jj0 * 3 + 2];
    const float p1x = sV[jj1 * 3 + 0], p1m = sV[jj1 * 3 + 1], p1y = sV[jj1 * 3 + 2];
    const float p2x = sV[(jj1 + 1) * 3 + 0], p2m = sV[(jj1 + 1) * 3 + 1], p2y = sV[(jj1 + 1) * 3 + 2];

    // vm[j] reads Vprev[j]; vx[j] reads Vprev[j+1]  (vectorized over j)
    const float vm0 = thm0 + lse3(Am0 + p0x, Am1 + p0m, Am2 + p0y);
    const float vx0 = thx0 + lse3(Ax0 + p1x, Ax1 + p1m, Ax2 + p1y);
    const float vm1 = thm1 + lse3(Am0 + p1x, Am1 + p1m, Am2 + p1y);
    const float vx1 = thx1 + lse3(Ax0 + p2x, Ax1 + p2m, Ax2 + p2y);

    __syncthreads();  // B1: all Vprev reads retired -> safe to overwrite in place

    sV[(jj0 + 1) * 3 + 0] = vx0; sV[(jj0 + 1) * 3 + 1] = vm0;
    sV[(jj1 + 1) * 3 + 0] = vx1; sV[(jj1 + 1) * 3 + 1] = vm1;
    if (tid == 0) { sV[0] = 0.0f; sV[1] = 0.0f; sV[2] = 0.0f; }  // V[i][0] = 0
    __syncthreads();  // B2: new x,m components visible

    // y-recurrence: vy[j] = logaddexp(c[j], g[j] + vy[j-1]); pairs (g,c) form a monoid
    const float lx = sV[jj0 * 3 + 0], lm = sV[jj0 * 3 + 1];  // left neighbor of jj0
    const float c0 = thy0 + lse2(Ay0 + lx,  Ay1 + lm);
    const float g0 = thy0 + Ay2;
    const float c1 = thy1 + lse2(Ay0 + vx0, Ay1 + vm0);
    const float g1 = thy1 + Ay2;

    // thread-local compose (g0,c0) then (g1,c1)
    float gT = g0 + g1;
    float cT = laddexp(c1, g1 + c0);

    // wave32 inclusive Hillis-Steele scan via lane permutes
#pragma unroll
    for (int d = 1; d < 32; d <<= 1) {
      float gp = __shfl_up(gT, d, 32);
      float cp = __shfl_up(cT, d, 32);
      if (lane >= d) { cT = laddexp(cT, gT + cp); gT += gp; }
    }
    if (lane == 31) { sWg[wid] = gT; sWc[wid] = cT; }
    __syncthreads();  // B3

    if (wid == 0) {   // wave 0 scans the 32 wave totals
      float wg = sWg[lane], wc = sWc[lane];
#pragma unroll
      for (int d = 1; d < 32; d <<= 1) {
        float gp = __shfl_up(wg, d, 32);
        float cp = __shfl_up(wc, d, 32);
        if (lane >= d) { wc = laddexp(wc, wg + cp); wg += gp; }
      }
      sWgI[lane] = wg; sWcI[lane] = wc;
    }
    __syncthreads();  // B4

    // exclusive prefix entering this thread = (waves < wid) ∘ (lanes < lane)
    float Gw = 0.0f, Cw = -INFINITY;
    if (wid > 0) { Gw = sWgI[wid - 1]; Cw = sWcI[wid - 1]; }
    float gEx = __shfl_up(gT, 1, 32);
    float cEx = __shfl_up(cT, 1, 32);
    if (lane == 0) { gEx = 0.0f; cEx = -INFINITY; }
    const float Eg = Gw + gEx;
    const float Ec = laddexp(cEx, gEx + Cw);
    const float w  = laddexp(Ec, Eg);           // carry = vy[jj0-1] (w_init = 0)
    const float vy0 = laddexp(c0, g0 + w);
    const float vy1 = laddexp(c1, g1 + vy0);

    sV[(jj0 + 1) * 3 + 2] = vy0;
    sV[(jj1 + 1) * 3 + 2] = vy1;

    // Prefetch theta row r+2 into the buffer just consumed (all reads done by B1<B4)
    if (tid < 32 && (r + 2) < NROWS) {
      tdm_load_row(theta + (size_t)(r + 2) * ROWELEMS,
                   (unsigned)(uintptr_t)&sTheta[r & 1][0]);
    }
  }

  __syncthreads();
  if (tid == 0) {
    out[0] = lse3(sV[MCOLS * 3 + 0], sV[MCOLS * 3 + 1], sV[MCOLS * 3 + 2]);
  }
}

extern "C" void kernel_launch(void* const* d_in, const int* in_sizes, int n_in,
                              void* d_out, int out_size, void* d_ws, size_t ws_size,
                              hipStream_t stream) {
  (void)in_sizes; (void)n_in; (void)out_size; (void)d_ws; (void)ws_size;
  const float* theta = (const float*)d_in[0];
  const float* A     = (const float*)d_in[1];
  float*       out   = (float*)d_out;
  viterbi_soft_kernel<<<dim3(1), dim3(1024), 0, stream>>>(theta, A, out);
}